// CorefHead_34565896798222
// MI455X (gfx1250) — compile-verified
//
#include <hip/hip_runtime.h>

// ---------------------------------------------------------------------------
// CorefHead on MI455X (gfx1250, wave32).
//   emb[768,768] = segmented-mean(bert[256,512,768], spans)     (memory-bound)
//   x  [768,512] = leaky_relu(emb @ W1 + b1)   -> V_WMMA_F32_16X16X4_F32
//   out[768,3]   = x @ W2 + b2
// Roofline: ~47MB of span reads @ 23.3 TB/s ~= 2us; 0.6 GFLOP GEMM is hidden.
// fp32 WMMA keeps reference precision at zero cost (compute is not the limit).
// ---------------------------------------------------------------------------

typedef __attribute__((ext_vector_type(2))) float v2f;   // A/B frag: 2 VGPRs
typedef __attribute__((ext_vector_type(8))) float v8f;   // C/D frag: 8 VGPRs

#define BB    256
#define SS    512
#define HH    768
#define EE    3
#define HID   512
#define NCLS  3
#define MM    (BB * EE)        // 768 pooled rows
#define ASTR  772              // LDS A-tile row stride (768 + 4 pad; 16B-aligned,
                               // 772 % 64 == 4 -> conflict-free half-wave reads)

// ---------------- Kernel 1: segmented mean pooling ------------------------
// One block per (b,e) row; lanes sweep H coalesced; inner loop walks the span.
__global__ void coref_pool_kernel(const float* __restrict__ bert,
                                  const int*   __restrict__ starts,
                                  const int*   __restrict__ ends,
                                  float*       __restrict__ emb) {
    const int row = blockIdx.x;                 // 0..767  (row = b*3 + e)
    const int s0  = starts[row];
    const int s1  = ends[row];
    const int cnt = s1 - s0;
    const float inv = (cnt > 0) ? (1.0f / (float)cnt) : 0.0f;  // (0,0) span -> 0
    const int b = row / EE;
    const float* __restrict__ base = bert + (size_t)b * SS * HH;
    for (int h = threadIdx.x; h < HH; h += blockDim.x) {
        float acc = 0.0f;
        for (int s = s0; s < s1; ++s)
            acc += base[(size_t)s * HH + h];
        emb[(size_t)row * HH + h] = acc * inv;
    }
}

// ---------------- Kernel 2: GEMM1 (+bias, LeakyReLU) via f32 WMMA ---------
// grid = (HID/64, MM/16); block = 128 threads = 4 waves.
// Each block: M-tile of 16 rows, N-chunk of 64 cols (one 16-wide tile/wave).
// A-tile (16x768 f32) staged in LDS once, reused by all 192 K-steps.
__global__ void coref_gemm1_kernel(const float* __restrict__ emb,
                                   const float* __restrict__ W1,
                                   const float* __restrict__ b1,
                                   float*       __restrict__ x) {
    __shared__ float As[16 * ASTR];

    const int tid  = threadIdx.x;
    const int wave = tid >> 5;                  // 0..3
    const int lane = tid & 31;
    const int row0 = blockIdx.y * 16;           // M-tile origin
    const int n0   = blockIdx.x * 64 + wave * 16;

    // Cooperative float4 stage of the contiguous 16x768 A block into padded LDS.
    const float* __restrict__ src = emb + (size_t)row0 * HH;
    for (int i = tid; i < (16 * HH) / 4; i += blockDim.x) {
        const int r = (i * 4) / HH;
        const int c = (i * 4) % HH;
        *(float4*)&As[r * ASTR + c] = *(const float4*)(src + (size_t)r * HH + c);
    }
    __syncthreads();

    // Fragment coordinates (ISA 7.12.2, wave32):
    //   A 16x4 f32 : lane L -> row M=L%16, Ks {k+2*(L>=16), k+2*(L>=16)+1}
    //   B  4x16 f32: lane L -> col N=L%16, Ks {k+2*(L>=16), k+2*(L>=16)+1}
    //   C/D 16x16  : VGPR i -> M = i + 8*(L>=16), N = L%16
    const int m  = lane & 15;
    const int hi = lane >> 4;
    const int n  = n0 + m;

    v8f acc = {0.f, 0.f, 0.f, 0.f, 0.f, 0.f, 0.f, 0.f};

    for (int k = 0; k < HH; k += 4) {
        const int kk = k + 2 * hi;
        v2f a = *(const v2f*)&As[m * ASTR + kk];          // ds_load_b64
        v2f b;
        b.x = W1[(size_t)kk * HID + n];                   // L2-resident (1.5MB)
        b.y = W1[(size_t)(kk + 1) * HID + n];
        // emits: v_wmma_f32_16x16x4_f32  (8-arg form: neg_a,A,neg_b,B,cmod,C,ra,rb)
        acc = __builtin_amdgcn_wmma_f32_16x16x4_f32(
                  false, a, false, b, (short)0, acc, false, false);
    }

    // Epilogue: bias + LeakyReLU(0.01), scatter per C/D layout.
    const float bias = b1[n];
#pragma unroll
    for (int i = 0; i < 8; ++i) {
        const int mrow = i + 8 * hi;
        float v = acc[i] + bias;
        v = (v > 0.0f) ? v : 0.01f * v;
        x[(size_t)(row0 + mrow) * HID + n] = v;
    }
}

// ---------------- Kernel 3: head GEMM2 (768x512 @ 512x3 + b2) -------------
// 2304 outputs; one 512-length dot per thread (negligible: 2.4 MFLOP, L2-hot).
__global__ void coref_head_kernel(const float* __restrict__ x,
                                  const float* __restrict__ W2,
                                  const float* __restrict__ b2,
                                  float*       __restrict__ out) {
    const int idx = blockIdx.x * blockDim.x + threadIdx.x;
    if (idx >= MM * NCLS) return;
    const int m = idx / NCLS;
    const int c = idx % NCLS;
    const float* __restrict__ xr = x + (size_t)m * HID;
    float acc = b2[c];
    for (int k = 0; k < HID; ++k)
        acc = fmaf(xr[k], W2[k * NCLS + c], acc);
    out[idx] = acc;
}

// ---------------------------------------------------------------------------
extern "C" void kernel_launch(void* const* d_in, const int* in_sizes, int n_in,
                              void* d_out, int out_size, void* d_ws, size_t ws_size,
                              hipStream_t stream) {
    const float* bert   = (const float*)d_in[0];   // [256,512,768] f32
    const int*   starts = (const int*)  d_in[1];   // [256,3] int
    const int*   ends   = (const int*)  d_in[2];   // [256,3] int
    const float* W1     = (const float*)d_in[3];   // [768,512] f32
    const float* b1     = (const float*)d_in[4];   // [512] f32
    const float* W2     = (const float*)d_in[5];   // [512,3] f32
    const float* b2     = (const float*)d_in[6];   // [3] f32
    float*       out    = (float*)d_out;           // [768,3] f32

    float* emb = (float*)d_ws;                     // 768*768 f32 = 2.25 MB
    float* x   = emb + (size_t)MM * HH;            // 768*512 f32 = 1.50 MB

    coref_pool_kernel<<<MM, 128, 0, stream>>>(bert, starts, ends, emb);
    coref_gemm1_kernel<<<dim3(HID / 64, MM / 16), 128, 0, stream>>>(emb, W1, b1, x);
    coref_head_kernel<<<(MM * NCLS + 255) / 256, 256, 0, stream>>>(x, W2, b2, out);
}